// GatedGCN_24541443129598
// MI455X (gfx1250) — compile-verified
//
#include <hip/hip_runtime.h>
#include <hip/hip_bf16.h>
#include <math.h>

#define N_NODES 50000
#define N_EDGES 800000
#define HID 128
#define G3 384          // 3*HID (GRU gate width)
#define N_ETYPES 4
#define N_LAYERS 2
#define N_STEPS 5
#define BN_EPS 1e-5f

typedef float v2f __attribute__((ext_vector_type(2)));
typedef float v8f __attribute__((ext_vector_type(8)));

// Full-precision f32 WMMA: D(16x16) = A(16x4) * B(4x16) + C
__device__ __forceinline__ v8f wmma4(v2f a, v2f b, v8f c) {
    return __builtin_amdgcn_wmma_f32_16x16x4_f32(
        /*neg_a=*/false, a, /*neg_b=*/false, b,
        /*c_mod=*/(short)0, c, /*reuse_a=*/false, /*reuse_b=*/false);
}

// ---------------------------------------------------------------------------
// All-etype message transform: H[et][n][f] = sum_d h[n][d]*Wmsg[et][f][d] + b
// One wave computes the same 16x16 (node,feat) tile for all 4 etypes,
// sharing the A (h) fragment: 4 WMMAs per A-load.
// grid = N/16, block = 256 (8 waves -> 8 feature tiles covering 128 cols)
// ---------------------------------------------------------------------------
__global__ void msg_gemm4(const float* __restrict__ h,
                          const float* __restrict__ Wm,   // [4][128][128] (layer slice)
                          const float* __restrict__ bm,   // [4][128]
                          float* __restrict__ H) {        // [4][N][128]
    const int wave = threadIdx.x >> 5;
    const int lane = threadIdx.x & 31;
    const int row0 = blockIdx.x << 4;

    const int mrow  = row0 + (lane & 15);
    const int fcol  = (wave << 4) + (lane & 15);
    const int khalf = (lane >> 4) << 1;            // 0 or 2

    v8f acc0 = {}, acc1 = {}, acc2 = {}, acc3 = {};
    const float* W0 = Wm;
    const float* W1 = Wm + 1 * HID * HID;
    const float* W2 = Wm + 2 * HID * HID;
    const float* W3 = Wm + 3 * HID * HID;

    for (int kc = 0; kc < HID; kc += 4) {
        const int kb = kc + khalf;
        v2f a;
        a.x = h[(size_t)mrow * HID + kb];
        a.y = h[(size_t)mrow * HID + kb + 1];
        v2f b0, b1, b2, b3;
        b0.x = W0[(size_t)fcol * HID + kb]; b0.y = W0[(size_t)fcol * HID + kb + 1];
        b1.x = W1[(size_t)fcol * HID + kb]; b1.y = W1[(size_t)fcol * HID + kb + 1];
        b2.x = W2[(size_t)fcol * HID + kb]; b2.y = W2[(size_t)fcol * HID + kb + 1];
        b3.x = W3[(size_t)fcol * HID + kb]; b3.y = W3[(size_t)fcol * HID + kb + 1];
        acc0 = wmma4(a, b0, acc0);
        acc1 = wmma4(a, b1, acc1);
        acc2 = wmma4(a, b2, acc2);
        acc3 = wmma4(a, b3, acc3);
    }
    const float bv0 = bm[fcol];
    const float bv1 = bm[HID + fcol];
    const float bv2 = bm[2 * HID + fcol];
    const float bv3 = bm[3 * HID + fcol];
    const int rbase = row0 + ((lane >> 4) << 3);
#pragma unroll
    for (int v = 0; v < 8; ++v) {
        const size_t ro = (size_t)(rbase + v) * HID + fcol;
        H[ro]                              = acc0[v] + bv0;
        H[(size_t)N_NODES * HID     + ro]  = acc1[v] + bv1;
        H[(size_t)N_NODES * HID * 2 + ro]  = acc2[v] + bv2;
        H[(size_t)N_NODES * HID * 3 + ro]  = acc3[v] + bv3;
    }
}

// ---------------------------------------------------------------------------
// gh = h @ Whh^T + bhh : each wave computes the r/z/n column blocks
// (fcol, fcol+128, fcol+256) sharing the A fragment: 3 WMMAs per A-load.
// grid = N/16, block = 256
// ---------------------------------------------------------------------------
__global__ void gh_gemm3(const float* __restrict__ h,
                         const float* __restrict__ Whh,   // [384][128]
                         const float* __restrict__ bhh,   // [384]
                         float* __restrict__ gh) {        // [N][384]
    const int wave = threadIdx.x >> 5;
    const int lane = threadIdx.x & 31;
    const int row0 = blockIdx.x << 4;

    const int mrow  = row0 + (lane & 15);
    const int fcol  = (wave << 4) + (lane & 15);
    const int khalf = (lane >> 4) << 1;

    v8f a0 = {}, a1 = {}, a2 = {};
    for (int kc = 0; kc < HID; kc += 4) {
        const int kb = kc + khalf;
        v2f a;
        a.x = h[(size_t)mrow * HID + kb];
        a.y = h[(size_t)mrow * HID + kb + 1];
        v2f b0, b1, b2;
        b0.x = Whh[(size_t)(fcol      ) * HID + kb];
        b0.y = Whh[(size_t)(fcol      ) * HID + kb + 1];
        b1.x = Whh[(size_t)(fcol + 128) * HID + kb];
        b1.y = Whh[(size_t)(fcol + 128) * HID + kb + 1];
        b2.x = Whh[(size_t)(fcol + 256) * HID + kb];
        b2.y = Whh[(size_t)(fcol + 256) * HID + kb + 1];
        a0 = wmma4(a, b0, a0);
        a1 = wmma4(a, b1, a1);
        a2 = wmma4(a, b2, a2);
    }
    const float b0_ = bhh[fcol];
    const float b1_ = bhh[fcol + 128];
    const float b2_ = bhh[fcol + 256];
    const int rbase = row0 + ((lane >> 4) << 3);
#pragma unroll
    for (int v = 0; v < 8; ++v) {
        const size_t ro = (size_t)(rbase + v) * G3;
        gh[ro + fcol]       = a0[v] + b0_;
        gh[ro + 128 + fcol] = a1[v] + b1_;
        gh[ro + 256 + fcol] = a2[v] + b2_;
    }
}

// ---------------------------------------------------------------------------
// Edge scatter: agg[dst] += H[etype*N + src]
// one thread per (edge, 4 contiguous feats)
// ---------------------------------------------------------------------------
__global__ void scatter_edges(const float* __restrict__ H,
                              const int* __restrict__ src,
                              const int* __restrict__ dst,
                              const int* __restrict__ ety,
                              float* __restrict__ agg) {
    const long long gid = (long long)blockIdx.x * blockDim.x + threadIdx.x;
    const int e = (int)(gid >> 5);
    if (e >= N_EDGES) return;
    const int q = ((int)gid & 31) << 2;
    const float4 m = *(const float4*)(H + ((size_t)ety[e] * N_NODES + src[e]) * HID + q);
    float* ap = agg + (size_t)dst[e] * HID + q;
    unsafeAtomicAdd(ap + 0, m.x);
    unsafeAtomicAdd(ap + 1, m.y);
    unsafeAtomicAdd(ap + 2, m.z);
    unsafeAtomicAdd(ap + 3, m.w);
}

// ---------------------------------------------------------------------------
// Fused gi GEMM (a @ Wih^T + bih, 3 gate blocks per wave) + GRU gates.
// grid = N/16, block = 256
// ---------------------------------------------------------------------------
__global__ void gru_fused(const float* __restrict__ agg,
                          const float* __restrict__ Wih,   // [384][128]
                          const float* __restrict__ bih,   // [384]
                          const float* __restrict__ gh,    // [N][384] (incl. bhh)
                          const float* __restrict__ hprev,
                          float* __restrict__ hnext) {
    const int wave = threadIdx.x >> 5;
    const int lane = threadIdx.x & 31;
    const int row0 = blockIdx.x << 4;

    const int mrow  = row0 + (lane & 15);
    const int fcol  = (wave << 4) + (lane & 15);
    const int khalf = (lane >> 4) << 1;

    v8f ar = {}, az = {}, an = {};
    for (int kc = 0; kc < HID; kc += 4) {
        const int kb = kc + khalf;
        v2f a;
        a.x = agg[(size_t)mrow * HID + kb];
        a.y = agg[(size_t)mrow * HID + kb + 1];
        v2f br, bz, bn;
        br.x = Wih[(size_t)(fcol      ) * HID + kb];
        br.y = Wih[(size_t)(fcol      ) * HID + kb + 1];
        bz.x = Wih[(size_t)(fcol + 128) * HID + kb];
        bz.y = Wih[(size_t)(fcol + 128) * HID + kb + 1];
        bn.x = Wih[(size_t)(fcol + 256) * HID + kb];
        bn.y = Wih[(size_t)(fcol + 256) * HID + kb + 1];
        ar = wmma4(a, br, ar);
        az = wmma4(a, bz, az);
        an = wmma4(a, bn, an);
    }
    const float br_ = bih[fcol];
    const float bz_ = bih[fcol + 128];
    const float bn_ = bih[fcol + 256];
    const int rbase = row0 + ((lane >> 4) << 3);
#pragma unroll
    for (int v = 0; v < 8; ++v) {
        const int r = rbase + v;
        const float ghr = gh[(size_t)r * G3 + fcol];
        const float ghz = gh[(size_t)r * G3 + 128 + fcol];
        const float ghn = gh[(size_t)r * G3 + 256 + fcol];
        const float hp  = hprev[(size_t)r * HID + fcol];
        const float rg = 1.f / (1.f + __expf(-(ar[v] + br_ + ghr)));
        const float zg = 1.f / (1.f + __expf(-(az[v] + bz_ + ghz)));
        const float ng = tanhf(an[v] + bn_ + rg * ghn);
        hnext[(size_t)r * HID + fcol] = (1.f - zg) * ng + zg * hp;
    }
}

// ---------------------------------------------------------------------------
__global__ void zero_kernel(float* __restrict__ p, long long n) {
    long long i = (long long)blockIdx.x * blockDim.x + threadIdx.x;
    const long long stride = (long long)gridDim.x * blockDim.x;
    for (; i < n; i += stride) p[i] = 0.f;
}

// stats[0..127] = sum(relu(h)), stats[128..255] = sum(relu(h)^2)
__global__ void bn_stats(const float* __restrict__ h, float* __restrict__ stats) {
    const int c = threadIdx.x & 127;
    const int rowsPerBlk = blockDim.x >> 7;
    int r = blockIdx.x * rowsPerBlk + (threadIdx.x >> 7);
    const int rstride = gridDim.x * rowsPerBlk;
    float s = 0.f, ss = 0.f;
    for (; r < N_NODES; r += rstride) {
        float v = h[(size_t)r * HID + c];
        v = v > 0.f ? v : 0.f;
        s += v; ss += v * v;
    }
    unsafeAtomicAdd(&stats[c], s);
    unsafeAtomicAdd(&stats[128 + c], ss);
}

__global__ void bn_norm(const float* __restrict__ h,
                        const float* __restrict__ stats,
                        const float* __restrict__ gamma,
                        const float* __restrict__ beta,
                        float* __restrict__ out, int col_off) {
    const long long gid = (long long)blockIdx.x * blockDim.x + threadIdx.x;
    if (gid >= (long long)N_NODES * HID) return;
    const int c = (int)(gid & 127);
    const long long r = gid >> 7;
    const float inv  = 1.f / (float)N_NODES;
    const float mean = stats[c] * inv;
    const float var  = stats[128 + c] * inv - mean * mean;
    float v = h[gid];
    v = v > 0.f ? v : 0.f;
    out[r * (HID * N_LAYERS) + col_off + c] =
        (v - mean) * rsqrtf(var + BN_EPS) * gamma[c] + beta[c];
}

// ---------------------------------------------------------------------------
extern "C" void kernel_launch(void* const* d_in, const int* in_sizes, int n_in,
                              void* d_out, int out_size, void* d_ws, size_t ws_size,
                              hipStream_t stream) {
    const float* x     = (const float*)d_in[0];
    const int*   src   = (const int*)d_in[1];
    const int*   dst   = (const int*)d_in[2];
    const int*   ety   = (const int*)d_in[3];
    const float* Wmsg  = (const float*)d_in[4];   // [2][4][128][128]
    const float* bmsg  = (const float*)d_in[5];   // [2][4][128]
    const float* Wih   = (const float*)d_in[6];   // [2][384][128]
    const float* Whh   = (const float*)d_in[7];   // [2][384][128]
    const float* bih   = (const float*)d_in[8];   // [2][384]
    const float* bhh   = (const float*)d_in[9];   // [2][384]
    const float* gamma = (const float*)d_in[10];  // [2][128]
    const float* beta  = (const float*)d_in[11];  // [2][128]
    float* out = (float*)d_out;
    float* ws  = (float*)d_ws;

    const size_t NH = (size_t)N_NODES * HID;      // 6.4M floats
    float* hA    = ws;                // [N][128]
    float* hB    = ws + NH;           // [N][128]
    float* agg   = ws + 2 * NH;       // [N][128]
    float* big   = ws + 3 * NH;       // 4*NH: H[4][N][128], reused as gh[N][384]
    float* stats = ws + 7 * NH;       // 256 floats

    const int ntiles = N_NODES / 16;  // 3125, exact

    for (int l = 0; l < N_LAYERS; ++l) {
        // h0 = x (IN_FEATS == HIDDEN, so the zero-pad is an identity)
        hipMemcpyAsync(hA, x, NH * sizeof(float), hipMemcpyDeviceToDevice, stream);
        float* hc = hA;
        float* hn = hB;

        const float* Wm_l  = Wmsg + (size_t)l * N_ETYPES * HID * HID;
        const float* bm_l  = bmsg + (size_t)l * N_ETYPES * HID;
        const float* Wih_l = Wih  + (size_t)l * G3 * HID;
        const float* Whh_l = Whh  + (size_t)l * G3 * HID;
        const float* bih_l = bih  + (size_t)l * G3;
        const float* bhh_l = bhh  + (size_t)l * G3;

        for (int s = 0; s < N_STEPS; ++s) {
            // H[et] = h @ Wmsg[et]^T + bmsg[et], all etypes per wave
            msg_gemm4<<<ntiles, 256, 0, stream>>>(hc, Wm_l, bm_l, big);
            // agg = segment_sum over edges
            zero_kernel<<<4096, 256, 0, stream>>>(agg, (long long)NH);
            scatter_edges<<<(N_EDGES * 32 + 255) / 256, 256, 0, stream>>>(big, src, dst, ety, agg);
            // gh = h @ Whh^T + bhh  (reuses `big`; H already consumed)
            gh_gemm3<<<ntiles, 256, 0, stream>>>(hc, Whh_l, bhh_l, big);
            // gi GEMM + GRU gates fused
            gru_fused<<<ntiles, 256, 0, stream>>>(agg, Wih_l, bih_l, big, hc, hn);
            float* t = hc; hc = hn; hn = t;
        }

        // ReLU + BatchNorm (training mode, biased variance) into out slice
        zero_kernel<<<1, 256, 0, stream>>>(stats, 256);
        bn_stats<<<512, 256, 0, stream>>>(hc, stats);
        bn_norm<<<(int)((NH + 255) / 256), 256, 0, stream>>>(
            hc, stats, gamma + l * HID, beta + l * HID, out, l * HID);
    }
}